// CTCLayer_14078902796901
// MI455X (gfx1250) — compile-verified
//
#include <hip/hip_runtime.h>
#include <hip/hip_bf16.h>

// Problem shape from the reference setup_inputs(): B=512, T=512, C=256, L=128.
#define TT 512
#define CC 256                  // classes; blank = CC-1
#define LL 128
#define SS (2 * LL + 1)         // 257 extended states
#define BLANKC (CC - 1)
#define NEGV (-1.0e30f)         // finite stand-in for log(0) (same in base-2 domain)
#define EPSV (1.0e-7f)
#define LN2F 0.6931471805599453f
#define NTHREADS 288            // 9 wave32s: waves 0-7 issue DMA, lane tid=256 owns s=256

// Raw hardware transcendentals (base-2): v_exp_f32 / v_log_f32, no scale muls.
__device__ __forceinline__ float hw_exp2(float x) { return __builtin_amdgcn_exp2f(x); }
__device__ __forceinline__ float hw_log2(float x) { return __builtin_amdgcn_logf(x); }

// CDNA5 async global->LDS copy (ASYNCcnt-tracked). GVS addressing:
// mem = SADDR(64b, SGPR pair) + VADDR(32b byte offset); LDS dest byte address in VGPR.
__device__ __forceinline__ void async_row_load(const float* __restrict__ base,
                                               float* lds_dst, unsigned byte_off) {
  unsigned lds_addr = (unsigned)(size_t)lds_dst;   // low 32 bits of shared ptr = LDS offset
  asm volatile("global_load_async_to_lds_b32 %0, %1, %2"
               :: "v"(lds_addr), "v"(byte_off), "s"(base)
               : "memory");
}

__global__ __launch_bounds__(NTHREADS) void ctc_fwd_kernel(
    const int* __restrict__ y_true,     // (B, L) int32
    const float* __restrict__ y_pred,   // (B, T, C) float32 (softmax probs)
    float* __restrict__ out)            // (B,) float32
{
  __shared__ float rowRaw[4][CC];         // quad-buffered raw prob rows (4 KB)
  __shared__ float alphaBuf[2][SS + 2];   // double-buffered alpha (base-2), 2-entry NEG pad

  const int tid = threadIdx.x;
  const int b = blockIdx.x;
  const float* base = y_pred + (size_t)b * (TT * CC);
  const int* lab = y_true + b * LL;

  // Per-thread extended label + skip flag (s = tid).
  int  myExt  = BLANKC;
  bool mySkip = false;
  if (tid < SS && (tid & 1)) {
    const int j = tid >> 1;               // (s-1)/2
    myExt = lab[j];
    if (tid >= 3) mySkip = (myExt != lab[j - 1]);
  }
  if (tid < 2) { alphaBuf[0][tid] = NEGV; alphaBuf[1][tid] = NEGV; }

  // Prologue: prefetch rows 0 and 1 (2 async ops in flight per issuing wave).
  if (tid < CC) {
    async_row_load(base, &rowRaw[0][tid], (unsigned)((0 * CC + tid) * 4));
    async_row_load(base, &rowRaw[1][tid], (unsigned)((1 * CC + tid) * 4));
  }

  // ---- t = 0 (alpha init), peeled out of the hot loop ----
  if (tid < CC)
    async_row_load(base, &rowRaw[2][tid], (unsigned)((2 * CC + tid) * 4));
  asm volatile("s_wait_asynccnt 2" ::: "memory");   // row 0 resident (in-order)
  __syncthreads();
  if (tid < SS) {
    const float lp2 = hw_log2(rowRaw[0][myExt] + EPSV);
    alphaBuf[1][tid + 2] = (tid < 2) ? lp2 : NEGV;  // alpha0 in base-2 domain
  }
  int cur = 1;

  // ---- steady state: t = 1 .. TT-1 ----
  for (int t = 1; t < TT; ++t) {
    // Issue row t+2 into buffer (t+2)&3. Safe: that buffer's last readers
    // (step t-2) are fenced off by the step t-1 barrier.
    if (t + 2 < TT && tid < CC)
      async_row_load(base, &rowRaw[(t + 2) & 3][tid],
                     (unsigned)(((t + 2) * CC + tid) * 4));

    // Async loads complete in order: <=2 outstanding newer rows => row t resident.
    if (t < TT - 2)       asm volatile("s_wait_asynccnt 2" ::: "memory");
    else if (t == TT - 2) asm volatile("s_wait_asynccnt 1" ::: "memory");
    else                  asm volatile("s_wait_asynccnt 0" ::: "memory");

    __syncthreads();  // all waves' row-t slices visible; also fences alpha buffers

    if (tid < SS) {
      const float lp2 = hw_log2(rowRaw[t & 3][myExt] + EPSV);
      const float a1 = alphaBuf[cur][tid + 2];                // stay
      const float a2 = alphaBuf[cur][tid + 1];                // from s-1 (pad -> NEG at s=0)
      const float a3 = mySkip ? alphaBuf[cur][tid] : NEGV;    // from s-2
      const float m  = fmaxf(a1, fmaxf(a2, a3));
      const float nw = m + hw_log2(hw_exp2(a1 - m) + hw_exp2(a2 - m) + hw_exp2(a3 - m)) + lp2;
      alphaBuf[cur ^ 1][tid + 2] = nw;
    }
    cur ^= 1;
  }

  __syncthreads();
  if (tid == 0) {
    const float x = alphaBuf[cur][SS + 1];   // alpha_T[S-1]  (base-2 domain)
    const float y = alphaBuf[cur][SS + 0];   // alpha_T[S-2]
    const float m = fmaxf(x, y);
    // loss = -ln(2) * log2addexp2(x, y): single ln2 scale at the very end
    out[b] = -LN2F * (m + hw_log2(hw_exp2(x - m) + hw_exp2(y - m)));
  }
}

extern "C" void kernel_launch(void* const* d_in, const int* in_sizes, int n_in,
                              void* d_out, int out_size, void* d_ws, size_t ws_size,
                              hipStream_t stream) {
  (void)in_sizes; (void)n_in; (void)d_ws; (void)ws_size;
  const int*   y_true = (const int*)d_in[0];    // (B, L) int32
  const float* y_pred = (const float*)d_in[1];  // (B, T, C) float32
  float* out = (float*)d_out;                   // (B, 1) float32 -> B floats
  const int B = out_size;                       // 512
  ctc_fwd_kernel<<<B, NTHREADS, 0, stream>>>(y_true, y_pred, out);
}